// GraphEncoder_83915071030119
// MI455X (gfx1250) — compile-verified
//
#include <hip/hip_runtime.h>
#include <math.h>

typedef float v2f __attribute__((ext_vector_type(2)));
typedef float v8f __attribute__((ext_vector_type(8)));

#define IN_CH  128
#define HID    128
#define OUT_CH 64
#define LDSPAD 132   // 128 + 4-float pad: A-frag reads hit 16 distinct LDS banks

// relaxed, device-scope, non-returning float atomic add -> global_atomic_add_f32
__device__ __forceinline__ void fadd_atomic(float* p, float v) {
  (void)__hip_atomic_fetch_add(p, v, __ATOMIC_RELAXED, __HIP_MEMORY_SCOPE_AGENT);
}

// ---------------- degree / normalization ----------------
__global__ void k_init_deg(int* deg, int n) {
  int i = blockIdx.x * blockDim.x + threadIdx.x;
  if (i < n) deg[i] = 1;                       // self-loop contributes 1
}
__global__ void k_count_deg(const int* __restrict__ dst, int* deg, int nE) {
  int e = blockIdx.x * blockDim.x + threadIdx.x;
  if (e < nE)
    (void)__hip_atomic_fetch_add(&deg[dst[e]], 1, __ATOMIC_RELAXED,
                                 __HIP_MEMORY_SCOPE_AGENT);
}
// NOTE: deg and dis alias the same buffer (per-element read-then-write) -> no __restrict__
__global__ void k_dis(const int* deg, float* dis, int n) {
  int i = blockIdx.x * blockDim.x + threadIdx.x;
  if (i < n) dis[i] = rsqrtf((float)deg[i]);
}

// ---------------- fuse Wmu|Wls into one 128x128 weight ----------------
__global__ void k_build_wcat(const float* __restrict__ Wmu,
                             const float* __restrict__ Wls,
                             float* __restrict__ Wcat) {
  int i = blockIdx.x * blockDim.x + threadIdx.x;   // 0..16383
  int k = i >> 7, j = i & 127;
  Wcat[i] = (j < OUT_CH) ? Wmu[k * OUT_CH + j] : Wls[k * OUT_CH + (j - OUT_CH)];
}

// ---------------- C[Mx128] = A[Mx128] @ B[128x128], row-major, M%16==0 -----
// grid.x = M/16 ; block = 256 (8 waves); wave w owns output cols [16w,16w+16)
__global__ void k_gemm128_wmma(const float* __restrict__ A,
                               const float* __restrict__ B,
                               float* __restrict__ C) {
  __shared__ float lds[16 * LDSPAD];
  const int tid = threadIdx.x;
  const int m0  = blockIdx.x * 16;

  // 16x128 A strip is 2048 contiguous floats; stage into padded LDS
  const float* Ab = A + (size_t)m0 * IN_CH;
  #pragma unroll
  for (int it = 0; it < 2; ++it) {
    int flat = (tid + it * 256) * 4;          // float4 granularity
    int r = flat >> 7, k = flat & 127;
    *(float4*)&lds[r * LDSPAD + k] = *(const float4*)&Ab[flat];
  }
  __syncthreads();

  const int lane = tid & 31;
  const int wave = tid >> 5;
  const int nr   = lane & 15;                 // row for A-frag, col for B/C
  const int half = lane >> 4;                 // lane half selects K pair / M+8
  const int col  = wave * 16 + nr;

  v8f c = {};
  #pragma unroll
  for (int kk = 0; kk < 128; kk += 4) {
    const int k0 = kk + 2 * half;
    // A 16x4 frag: lanes0-15 hold K={kk,kk+1}, lanes16-31 K={kk+2,kk+3}
    v2f a = *(const v2f*)&lds[nr * LDSPAD + k0];
    // B 4x16 frag: same K split, column striped across lanes
    v2f b;
    b.x = B[(size_t)k0       * HID + col];
    b.y = B[(size_t)(k0 + 1) * HID + col];
    c = __builtin_amdgcn_wmma_f32_16x16x4_f32(false, a, false, b,
                                              (short)0, c, false, false);
  }

  // C/D layout: VGPR r -> row r (lanes0-15) / row r+8 (lanes16-31)
  float* Cb = C + (size_t)m0 * HID + col;
  #pragma unroll
  for (int r = 0; r < 8; ++r)
    Cb[(size_t)(r + 8 * half) * HID] = c[r];
}

// -------- out[i*ldo+c] = feat[i*ldf+coff+c]*dis[i]^2 + bias[c] (self-loop) --
__global__ void k_self_bias(float* __restrict__ out, int ldo,
                            const float* __restrict__ feat, int ldf, int coff,
                            const float* __restrict__ dis,
                            const float* __restrict__ bias,
                            int nch, int n) {
  int t = blockIdx.x * blockDim.x + threadIdx.x;
  int per = nch >> 2;
  if (t >= n * per) return;
  int i = t / per;
  int c = (t % per) * 4;
  float d  = dis[i];
  float d2 = d * d;
  float4 f  = *(const float4*)&feat[(size_t)i * ldf + coff + c];
  float4 bv = *(const float4*)&bias[c];
  float4 o  = { f.x * d2 + bv.x, f.y * d2 + bv.y,
                f.z * d2 + bv.z, f.w * d2 + bv.w };
  *(float4*)&out[(size_t)i * ldo + c] = o;
}

// -------- conv1 edge scatter: h[dst][c] += h0[src][c] * dis[s]*dis[d] -------
// 32 threads per edge (128 ch / float4); gathers coalesced, atomics hit L2
__global__ void k_edge_scatter(const int* __restrict__ src,
                               const int* __restrict__ dst,
                               const float* __restrict__ dis,
                               const float* __restrict__ feat,
                               float* __restrict__ out,
                               int nE) {
  int t = blockIdx.x * blockDim.x + threadIdx.x;
  long long total = (long long)nE * 32;
  if ((long long)t >= total) return;
  int e = t >> 5;
  int c = (t & 31) * 4;
  int s = src[e], d = dst[e];
  float norm = dis[s] * dis[d];
  float4 f = *(const float4*)&feat[(size_t)s * HID + c];
  float* o = &out[(size_t)d * HID + c];
  fadd_atomic(o + 0, f.x * norm);
  fadd_atomic(o + 1, f.y * norm);
  fadd_atomic(o + 2, f.z * norm);
  fadd_atomic(o + 3, f.w * norm);
}

// -------- fused conv2/3 scatter: cols 0..63 -> mu, 64..127 -> logstd --------
// single pass over edges: one src/dst/dis gather feeds both outputs
__global__ void k_edge_scatter2(const int* __restrict__ src,
                                const int* __restrict__ dst,
                                const float* __restrict__ dis,
                                const float* __restrict__ feat,   // ld = 128
                                float* __restrict__ out_mu,       // ld = 64
                                float* __restrict__ out_ls,       // ld = 64
                                int nE) {
  int t = blockIdx.x * blockDim.x + threadIdx.x;
  long long total = (long long)nE * 32;
  if ((long long)t >= total) return;
  int e = t >> 5;
  int c = (t & 31) * 4;                       // 0..124 over fused channels
  int s = src[e], d = dst[e];
  float norm = dis[s] * dis[d];
  float4 f = *(const float4*)&feat[(size_t)s * HID + c];
  float* o = (c < OUT_CH) ? &out_mu[(size_t)d * OUT_CH + c]
                          : &out_ls[(size_t)d * OUT_CH + (c - OUT_CH)];
  fadd_atomic(o + 0, f.x * norm);
  fadd_atomic(o + 1, f.y * norm);
  fadd_atomic(o + 2, f.z * norm);
  fadd_atomic(o + 3, f.w * norm);
}

// -------- z = mu + eps * exp(logstd) ----------------------------------------
__global__ void k_z(float* __restrict__ z, const float* __restrict__ mu,
                    const float* __restrict__ ls, const float* __restrict__ eps,
                    int n) {
  int i = blockIdx.x * blockDim.x + threadIdx.x;
  if (i < n) z[i] = mu[i] + eps[i] * expf(ls[i]);
}

extern "C" void kernel_launch(void* const* d_in, const int* in_sizes, int n_in,
                              void* d_out, int out_size, void* d_ws, size_t ws_size,
                              hipStream_t stream) {
  const float* x   = (const float*)d_in[0];
  const int*   ei  = (const int*)d_in[1];     // edge_index [2,E] as int32
  const float* W1  = (const float*)d_in[2];
  const float* b1  = (const float*)d_in[3];
  const float* Wmu = (const float*)d_in[4];
  const float* bmu = (const float*)d_in[5];
  const float* Wls = (const float*)d_in[6];
  const float* bls = (const float*)d_in[7];
  const float* eps = (const float*)d_in[8];

  const int N = in_sizes[0] / IN_CH;          // 100000 (divisible by 16)
  const int E = in_sizes[1] / 2;              // 1600000
  const int* srcI = ei;
  const int* dstI = ei + E;

  float* out = (float*)d_out;
  float* z  = out;
  float* mu = out + (size_t)N * OUT_CH;
  float* ls = out + 2 * (size_t)N * OUT_CH;

  // workspace layout (floats): dis/deg | h0 (N*128) | h (N*128) | Wcat (16K)
  float* ws   = (float*)d_ws;
  float* dis  = ws;
  int*   degi = (int*)ws;
  size_t off  = (size_t)((N + 127) & ~127);
  float* h0   = ws + off;                     // also reused as fused mu|ls pre-act
  float* h    = h0 + (size_t)N * HID;
  float* wcat = h  + (size_t)N * HID;

  // --- degree + symmetric norm ---
  k_init_deg <<<(N + 255) / 256, 256, 0, stream>>>(degi, N);
  k_count_deg<<<(E + 255) / 256, 256, 0, stream>>>(dstI, degi, E);
  k_dis      <<<(N + 255) / 256, 256, 0, stream>>>(degi, dis, N);
  k_build_wcat<<<(HID * HID) / 256, 256, 0, stream>>>(Wmu, Wls, wcat);

  const long long et  = (long long)E * 32;
  const int eblocks   = (int)((et + 255) / 256);

  // --- conv1: h0 = x @ W1 ; h = scatter(h0) + self + b1 ---
  k_gemm128_wmma<<<N / 16, 256, 0, stream>>>(x, W1, h0);
  {
    int tot = N * (HID / 4);
    k_self_bias<<<(tot + 255) / 256, 256, 0, stream>>>(h, HID, h0, HID, 0, dis, b1, HID, N);
    k_edge_scatter<<<eblocks, 256, 0, stream>>>(srcI, dstI, dis, h0, h, E);
  }

  // --- conv2/3 fused: h0 <- h @ [Wmu|Wls] ---
  k_gemm128_wmma<<<N / 16, 256, 0, stream>>>(h, wcat, h0);
  {
    int tot = N * (OUT_CH / 4);
    k_self_bias<<<(tot + 255) / 256, 256, 0, stream>>>(mu, OUT_CH, h0, HID, 0, dis, bmu, OUT_CH, N);
    k_self_bias<<<(tot + 255) / 256, 256, 0, stream>>>(ls, OUT_CH, h0, HID, OUT_CH, dis, bls, OUT_CH, N);
    k_edge_scatter2<<<eblocks, 256, 0, stream>>>(srcI, dstI, dis, h0, mu, ls, E);
  }

  // --- reparametrization ---
  int n64 = N * OUT_CH;
  k_z<<<(n64 + 255) / 256, 256, 0, stream>>>(z, mu, ls, eps, n64);
}